// GATLayer_82935818485897
// MI455X (gfx1250) — compile-verified
//
#include <hip/hip_runtime.h>
#include <math.h>

#define N_NODES 50000
#define N_EDGES 1600000
#define IN_DIM 256
#define OUT_DIM 128
#define NEG_SLOPE 0.2f
#define N_STRIPS (N_NODES / 16)   // 3125 exact

typedef __attribute__((ext_vector_type(2))) float v2f;
typedef __attribute__((ext_vector_type(8))) float v8f;

// -------- float atomic max via native int atomics (no CAS loop) --------
__device__ __forceinline__ void atomicMaxF32(float* addr, float val) {
    if (val >= 0.0f)
        atomicMax((int*)addr, __float_as_int(val));
    else
        atomicMin((unsigned int*)addr, (unsigned int)__float_as_int(val));
}

// =======================================================================
// Kernel 1: h = x @ W + bW   via V_WMMA_F32_16X16X4_F32 (fp32 matrix pipe)
// One wave32 per 16-row strip; 8 column tiles of 16 -> 16x128 strip.
// A-frag (16x4 f32): lanes 0-15 hold K={k,k+1}, lanes 16-31 hold K={k+2,k+3}.
// B-frag (4x16 f32): VGPR0 = row k (+2 for hi lanes), VGPR1 = row k+1 (+2).
// C/D (16x16 f32):   VGPR v, lanes 0-15 -> M=v, lanes 16-31 -> M=v+8.
// =======================================================================
__global__ __launch_bounds__(128) void gat_gemm(const float* __restrict__ x,
                                                const float* __restrict__ W,
                                                const float* __restrict__ bW,
                                                float* __restrict__ h) {
    const int lane  = threadIdx.x & 31;
    const int wave  = threadIdx.x >> 5;
    const int strip = blockIdx.x * 4 + wave;
    if (strip >= N_STRIPS) return;          // wave-uniform: EXEC stays all-1s

    const int half = lane >> 4;             // 0 or 1 (K sub-pair)
    const int l15  = lane & 15;
    const int row  = strip * 16 + l15;
    const float* __restrict__ xrow = x + (size_t)row * IN_DIM + half * 2;

    v8f acc[8] = {};

    for (int k = 0; k < IN_DIM; k += 4) {
        v2f a;
        a.x = xrow[k];
        a.y = xrow[k + 1];
        const float* __restrict__ wbase = W + (size_t)(k + 2 * half) * OUT_DIM + l15;
#pragma unroll
        for (int n = 0; n < 8; ++n) {
            v2f b;
            b.x = wbase[n * 16];
            b.y = wbase[n * 16 + OUT_DIM];
            acc[n] = __builtin_amdgcn_wmma_f32_16x16x4_f32(
                false, a, false, b, (short)0, acc[n], false, false);
        }
    }

    // Writeback + bias: acc[n][v] -> h[strip*16 + v + 8*half][n*16 + l15]
#pragma unroll
    for (int n = 0; n < 8; ++n) {
        const int col   = n * 16 + l15;
        const float bias = bW[col];
        float* __restrict__ o = h + (size_t)(strip * 16 + 8 * half) * OUT_DIM + col;
#pragma unroll
        for (int v = 0; v < 8; ++v) {
            o[(size_t)v * OUT_DIM] = acc[n][v] + bias;
        }
    }
}

// =======================================================================
// Kernel 2: per-node scores s_src/s_dst = h[i] . a, plus init of the
// segment-softmax accumulators and zeroing the output accumulator.
// One wave32 per node; 4 features per lane; shfl_xor tree reduce.
// =======================================================================
__global__ __launch_bounds__(256) void gat_scores(const float* __restrict__ h,
                                                  const float* __restrict__ a,
                                                  float* __restrict__ s_src,
                                                  float* __restrict__ s_dst,
                                                  float* __restrict__ rmax,
                                                  float* __restrict__ denom,
                                                  float* __restrict__ out) {
    const int lane = threadIdx.x & 31;
    const int node = (blockIdx.x * blockDim.x + threadIdx.x) >> 5;
    if (node >= N_NODES) return;

    const float* __restrict__ hr = h + (size_t)node * OUT_DIM;
    float ss = 0.0f, sd = 0.0f;
#pragma unroll
    for (int j = 0; j < 4; ++j) {
        const int c = lane + 32 * j;
        const float v = hr[c];
        ss += v * a[c];
        sd += v * a[OUT_DIM + c];
    }
#pragma unroll
    for (int off = 16; off > 0; off >>= 1) {
        ss += __shfl_xor(ss, off, 32);
        sd += __shfl_xor(sd, off, 32);
    }
    if (lane == 0) {
        s_src[node] = ss;
        s_dst[node] = sd;
        rmax[node]  = __int_as_float(0xFF800000);   // -inf
        denom[node] = 0.0f;
    }
    // zero output accumulator row: 32 lanes x float4 = 128 floats
    float4* __restrict__ orow = (float4*)(out + (size_t)node * OUT_DIM);
    orow[lane] = make_float4(0.0f, 0.0f, 0.0f, 0.0f);
}

// =======================================================================
// Kernel 3: per-edge logits + leaky ReLU + segment max (atomic, per src)
// =======================================================================
__global__ __launch_bounds__(256) void gat_edge_max(const int* __restrict__ src,
                                                    const int* __restrict__ dst,
                                                    const float* __restrict__ s_src,
                                                    const float* __restrict__ s_dst,
                                                    const float* __restrict__ ba,
                                                    float* __restrict__ e_buf,
                                                    float* __restrict__ rmax) {
    const int i = blockIdx.x * blockDim.x + threadIdx.x;
    if (i >= N_EDGES) return;
    const int s = src[i];
    const int d = dst[i];
    float e = s_src[s] + s_dst[d] + ba[0];
    e = (e >= 0.0f) ? e : NEG_SLOPE * e;
    e_buf[i] = e;
    atomicMaxF32(&rmax[s], e);
}

// =======================================================================
// Kernel 4: ex = exp(e - rowmax[src]); denom[src] += ex (atomic)
// =======================================================================
__global__ __launch_bounds__(256) void gat_edge_exp(const int* __restrict__ src,
                                                    const float* __restrict__ rmax,
                                                    float* __restrict__ e_buf,
                                                    float* __restrict__ denom) {
    const int i = blockIdx.x * blockDim.x + threadIdx.x;
    if (i >= N_EDGES) return;
    const int s = src[i];
    const float ex = __expf(e_buf[i] - rmax[s]);
    e_buf[i] = ex;
    atomicAdd(&denom[s], ex);
}

// =======================================================================
// Kernel 5: h'[src] += (ex/denom[src]) * h[dst]
// One wave32 per edge: 32 lanes x float4 = 128-feature row. Gathers and
// atomics stay resident in the 192MB L2 (h is 25.6MB).
// =======================================================================
__global__ __launch_bounds__(256) void gat_aggregate(const int* __restrict__ src,
                                                     const int* __restrict__ dst,
                                                     const float* __restrict__ e_buf,
                                                     const float* __restrict__ denom,
                                                     const float* __restrict__ h,
                                                     float* __restrict__ out) {
    const int lane = threadIdx.x & 31;
    const long tid = (long)blockIdx.x * blockDim.x + threadIdx.x;
    const int edge = (int)(tid >> 5);
    if (edge >= N_EDGES) return;

    const int s = src[edge];
    const int d = dst[edge];
    const float alpha = e_buf[edge] / denom[s];

    const float4 hv = ((const float4*)(h + (size_t)d * OUT_DIM))[lane];
    float* __restrict__ o = out + (size_t)s * OUT_DIM + lane * 4;
    atomicAdd(o + 0, alpha * hv.x);
    atomicAdd(o + 1, alpha * hv.y);
    atomicAdd(o + 2, alpha * hv.z);
    atomicAdd(o + 3, alpha * hv.w);
}

// =======================================================================
// Kernel 6: in-place ELU
// =======================================================================
__global__ __launch_bounds__(256) void gat_elu(float* __restrict__ out) {
    const long i = (long)blockIdx.x * blockDim.x + threadIdx.x;
    if (i >= (long)N_NODES * OUT_DIM) return;
    const float v = out[i];
    out[i] = (v > 0.0f) ? v : (__expf(v) - 1.0f);
}

// =======================================================================
extern "C" void kernel_launch(void* const* d_in, const int* in_sizes, int n_in,
                              void* d_out, int out_size, void* d_ws, size_t ws_size,
                              hipStream_t stream) {
    const float* x    = (const float*)d_in[0];
    const int*   eidx = (const int*)d_in[1];
    const float* W    = (const float*)d_in[2];
    const float* bW   = (const float*)d_in[3];
    const float* a    = (const float*)d_in[4];
    const float* ba   = (const float*)d_in[5];
    float* out = (float*)d_out;

    // workspace layout (floats)
    float* ws     = (float*)d_ws;
    float* h      = ws;                                   // 50000*128 = 6.4M
    float* s_src  = h + (size_t)N_NODES * OUT_DIM;        // 50k
    float* s_dst  = s_src + N_NODES;                      // 50k
    float* rmax   = s_dst + N_NODES;                      // 50k
    float* denom  = rmax + N_NODES;                       // 50k
    float* e_buf  = denom + N_NODES;                      // 1.6M

    const int* src = eidx;            // edge_index[0]
    const int* dst = eidx + N_EDGES;  // edge_index[1]

    gat_gemm<<<(N_STRIPS + 3) / 4, 128, 0, stream>>>(x, W, bW, h);
    gat_scores<<<(N_NODES * 32 + 255) / 256, 256, 0, stream>>>(
        h, a, s_src, s_dst, rmax, denom, out);
    gat_edge_max<<<(N_EDGES + 255) / 256, 256, 0, stream>>>(
        src, dst, s_src, s_dst, ba, e_buf, rmax);
    gat_edge_exp<<<(N_EDGES + 255) / 256, 256, 0, stream>>>(
        src, rmax, e_buf, denom);
    gat_aggregate<<<(int)(((long)N_EDGES * 32) / 256), 256, 0, stream>>>(
        src, dst, e_buf, denom, h, out);
    gat_elu<<<((long)N_NODES * OUT_DIM + 255) / 256, 256, 0, stream>>>(out);
}